// CMaxPool4d_87084756893727
// MI455X (gfx1250) — compile-verified
//
#include <hip/hip_runtime.h>
#include <stdint.h>
#include <math.h>

// Input  (8, 2, 32, 16, 16, 16) fp32 — float strides:
#define B_STR    262144
#define C_STR    131072
#define D1_STR   4096
#define D2_STR   256
#define D3_STR   16
// Output (8, 2, 16, 8, 8, 8) fp32 — float strides:
#define OB_STR   16384
#define OC_STR   8192
#define O1_STR   512
#define O2_STR   64
#define O3_STR   8

__global__ __launch_bounds__(256)
void cmaxpool4d_kernel(const float* __restrict__ in, float* __restrict__ out) {
    // 64 KiB LDS: 4 rows of 4096 floats, row cc = c*2 + w1
    __shared__ __align__(16) float smem[16384];

    const int tid = threadIdx.x;
    const int b   = blockIdx.x >> 4;   // 8 batches
    const int o1  = blockIdx.x & 15;   // 16 pooled d1 positions

    const float* gb = in + (size_t)b * B_STR;

    // ---- Stage 4 contiguous 16 KiB rows into LDS via CDNA5 async DMA ----
    // Each row (c, d1=2*o1+w1) is 4096 contiguous floats; 256 threads x 4
    // iterations of 16B/lane -> fully coalesced global_load_async_to_lds_b128.
#pragma unroll
    for (int cc = 0; cc < 4; ++cc) {
        const int c  = cc >> 1;
        const int w1 = cc & 1;
        const float* src = gb + c * C_STR + (2 * o1 + w1) * D1_STR;
#pragma unroll
        for (int it = 0; it < 4; ++it) {
            const int e = (it * 256 + tid) * 4;   // float index (16B granule)
            const uint64_t gaddr = (uint64_t)(uintptr_t)(src + e);
            // low 32 bits of the flat shared address == LDS byte offset
            const uint32_t laddr = (uint32_t)(uintptr_t)(&smem[cc * 4096 + e]);
            asm volatile("global_load_async_to_lds_b128 %0, %1, off"
                         :: "v"(laddr), "v"(gaddr)
                         : "memory");
        }
    }
    // Wait for this wave's async DMA, then sync so every wave's rows are visible.
    asm volatile("s_wait_asynccnt 0" ::: "memory");
    __syncthreads();

    // ---- Each thread: one o4-pair (two outputs) for both channels ----
    const int p4 = tid & 3;          // o4 pair: outputs 2*p4, 2*p4+1 (d4 base 4*p4)
    const int o3 = (tid >> 2) & 7;
    const int o2 = tid >> 5;

    float bm0 = -INFINITY, br0 = 0.f, bi0 = 0.f;
    float bm1 = -INFINITY, br1 = 0.f, bi1 = 0.f;

#pragma unroll
    for (int w1 = 0; w1 < 2; ++w1) {
#pragma unroll
        for (int w2 = 0; w2 < 2; ++w2) {
#pragma unroll
            for (int w3 = 0; w3 < 2; ++w3) {
                const int idx = (2 * o2 + w2) * D2_STR + (2 * o3 + w3) * D3_STR + 4 * p4;
                const float4 rv = *(const float4*)&smem[w1 * 4096 + idx];        // c = 0
                const float4 iv = *(const float4*)&smem[(2 + w1) * 4096 + idx];  // c = 1
                // Window flat index (w1,w2,w3,w4), w4 fastest; strict '>' keeps
                // the first maximum (JAX argmax semantics).
                float m;
                m = sqrtf(rv.x * rv.x + iv.x * iv.x);
                if (m > bm0) { bm0 = m; br0 = rv.x; bi0 = iv.x; }
                m = sqrtf(rv.y * rv.y + iv.y * iv.y);
                if (m > bm0) { bm0 = m; br0 = rv.y; bi0 = iv.y; }
                m = sqrtf(rv.z * rv.z + iv.z * iv.z);
                if (m > bm1) { bm1 = m; br1 = rv.z; bi1 = iv.z; }
                m = sqrtf(rv.w * rv.w + iv.w * iv.w);
                if (m > bm1) { bm1 = m; br1 = rv.w; bi1 = iv.w; }
            }
        }
    }

    const int ob = b * OB_STR + o1 * O1_STR + o2 * O2_STR + o3 * O3_STR + 2 * p4;
    float2 rr; rr.x = br0; rr.y = br1;
    float2 ii; ii.x = bi0; ii.y = bi1;
    *(float2*)&out[ob]          = rr;   // real channel
    *(float2*)&out[ob + OC_STR] = ii;   // imag channel
}

extern "C" void kernel_launch(void* const* d_in, const int* in_sizes, int n_in,
                              void* d_out, int out_size, void* d_ws, size_t ws_size,
                              hipStream_t stream) {
    (void)in_sizes; (void)n_in; (void)out_size; (void)d_ws; (void)ws_size;
    const float* in = (const float*)d_in[0];
    float* out = (float*)d_out;
    // 128 blocks = 8 batches x 16 pooled-d1 slices; 256 threads = 8 waves.
    cmaxpool4d_kernel<<<dim3(128), dim3(256), 0, stream>>>(in, out);
}